// ClusterMI_54477365182878
// MI455X (gfx1250) — compile-verified
//
#include <hip/hip_runtime.h>
#include <math.h>

// ---------------------------------------------------------------------------
// ClusterMI (KSG-style MI estimator) for MI455X / gfx1250.
// X @ X^T via V_WMMA_F32_16X16X4_F32 (full fp32 precision for k-NN ranks),
// fully fused (never materializes the 256MB distance matrix).
//
// Block = 8 wave32s, each wave owns a 16-row strip (block tile M=128).
// All waves share column stages: 32 columns (16.5KB) are staged into an LDS
// double buffer with GLOBAL_LOAD_ASYNC_TO_LDS_B128 (ASYNCcnt-tracked, no
// VGPR round trip) -> 8x less global/L2 traffic than per-wave streaming.
// WMMA B fragments are read with conflict-free ds_load_b64 (132-float
// padded stride). Top-4 selection and radius tests run in u32 space (bit
// pattern of non-negative floats is order-isomorphic, avoids v_max_num
// canonicalization); the cross-lane top-4 merge is a shfl_xor merge network.
// ---------------------------------------------------------------------------

typedef __attribute__((ext_vector_type(2))) float v2f;
typedef __attribute__((ext_vector_type(8))) float v8f;

#define N_PTS      8192
#define N_DIM      128
#define NCLS       10
#define KNN        3
#define STAGE_COLS 32
#define NSTAGE     (N_PTS / STAGE_COLS)   // 256
#define LDS_STRIDE 132                    // 128 + 4 pad floats: bank-conflict-free

__device__ __forceinline__ unsigned umin32(unsigned a, unsigned b) { return a < b ? a : b; }
__device__ __forceinline__ unsigned umax32(unsigned a, unsigned b) { return a > b ? a : b; }

__device__ __forceinline__ float digammaf_dev(float x) {
  if (x < 0.5f) x = 0.5f;                 // guard (m_i >= 3 in practice)
  float r = 0.f;
  while (x < 6.f) { r -= 1.f / x; x += 1.f; }
  float inv  = 1.f / x;
  float inv2 = inv * inv;
  return r + logf(x) - 0.5f * inv
       - inv2 * (1.f/12.f - inv2 * (1.f/120.f - inv2 * (1.f/252.f)));
}

// sorted insert keeping the 4 smallest (s0<=s1<=s2<=s3), u32 order == f32 order
__device__ __forceinline__ void insert4u(unsigned v, unsigned& s0, unsigned& s1,
                                         unsigned& s2, unsigned& s3) {
  unsigned n0 = umin32(s0, v);  unsigned x1 = umax32(s0, v);
  unsigned n1 = umin32(s1, x1); unsigned x2 = umax32(s1, x1);
  unsigned n2 = umin32(s2, x2); unsigned x3 = umax32(s2, x2);
  unsigned n3 = umin32(s3, x3);
  s0 = n0; s1 = n1; s2 = n2; s3 = n3;
}

__global__ void zero_counts_kernel(int* cnts) {
  if (threadIdx.x < NCLS) cnts[threadIdx.x] = 0;
}

// per-row squared norms + class histogram (integer atomics => deterministic)
__global__ __launch_bounds__(256) void prep_kernel(
    const float* __restrict__ X, const int* __restrict__ y,
    float* __restrict__ sqn, int* __restrict__ cnts) {
  int i = blockIdx.x * blockDim.x + threadIdx.x;
  if (i >= N_PTS) return;
  const float4* row = (const float4*)(X + (size_t)i * N_DIM);
  float s = 0.f;
#pragma unroll
  for (int k = 0; k < N_DIM / 4; ++k) {
    float4 v = row[k];
    s += v.x * v.x + v.y * v.y + v.z * v.z + v.w * v.w;
  }
  sqn[i] = s;
  atomicAdd(&cnts[y[i]], 1);
}

// 64 blocks x 256 threads; wave w of block b owns rows [(8b+w)*16, +16)
__global__ __launch_bounds__(256) void mi_main_kernel(
    const float* __restrict__ X, const int* __restrict__ y,
    const float* __restrict__ sqn, float* __restrict__ partials) {
  // double-buffered B stage: 2 x 32 cols x (128+4 pad) floats = 33KB
  __shared__ float bufB[2][STAGE_COLS][LDS_STRIDE];

  const unsigned HIGH_U = __float_as_uint(1e14f);   // (1e7)^2 class mask
  const unsigned BIG_U  = 0x7f7fffffu;              // FLT_MAX bits

  const int tid       = threadIdx.x;
  const int lane      = tid & 31;
  const int waveInBlk = tid >> 5;
  const int waveId    = (blockIdx.x * blockDim.x + tid) >> 5;
  const int rowBase   = waveId * 16;
  const int halfSel   = lane >> 4;     // 0 or 1
  const int lane15    = lane & 15;
  const int koff      = halfSel * 2;   // K sub-offset per WMMA f32 A/B layout

  // ---- preload A fragments for the 16-row strip (128 dims = 32 k-steps) ----
  v2f afrag[32];
  const float* arow = X + (size_t)(rowBase + lane15) * N_DIM + koff;
#pragma unroll
  for (int k = 0; k < 32; ++k)
    afrag[k] = *(const v2f*)(arow + 4 * k);

  // per-lane row metadata: this lane's C rows are rowBase + r + 8*halfSel
  float sqr[8];
  int   yr[8];
#pragma unroll
  for (int r = 0; r < 8; ++r) {
    int rowIdx = rowBase + r + 8 * halfSel;
    sqr[r] = sqn[rowIdx];
    yr[r]  = y[rowIdx];
  }

  // issue the async copy of one 32-column stage into LDS buffer b.
  // per wave: 4 x GLOBAL_LOAD_ASYNC_TO_LDS_B128; instr i moves column
  // (waveInBlk*4 + i), each lane carrying one 16B chunk (512B linear burst).
  auto issue_stage = [&](int s, int b) {
#pragma unroll
    for (int i = 0; i < 4; ++i) {
      const int col = waveInBlk * 4 + i;
      const float* gsrc = X + ((size_t)(s * STAGE_COLS + col) * N_DIM) + lane * 4;
      unsigned ldst = (unsigned)(size_t)(void*)&bufB[b][col][lane * 4];
      asm volatile("global_load_async_to_lds_b128 %0, %1, off"
                   :: "v"(ldst), "v"(gsrc) : "memory");
    }
  };

  // one 16x16 Gram tile from the LDS stage: 32 x (ds_load_b64 + wmma),
  // batched 16-at-a-time so DS loads pipeline ahead of the WMMA chain.
  auto tile_gemm = [&](int cur, int tt) -> v8f {
    v8f acc = {};
#pragma unroll
    for (int h = 0; h < 2; ++h) {
      v2f bf[16];
#pragma unroll
      for (int k = 0; k < 16; ++k)
        bf[k] = *(const v2f*)&bufB[cur][tt * 16 + lane15][(h * 16 + k) * 4 + koff];
#pragma unroll
      for (int k = 0; k < 16; ++k)
        acc = __builtin_amdgcn_wmma_f32_16x16x4_f32(
            false, afrag[h * 16 + k], false, bf[k], (short)0, acc, false, false);
    }
    return acc;
  };

  // =============== Pass A: 4 smallest same-class d^2 per row ===============
  unsigned s0[8], s1[8], s2[8], s3[8];
#pragma unroll
  for (int r = 0; r < 8; ++r) { s0[r] = s1[r] = s2[r] = s3[r] = BIG_U; }

  issue_stage(0, 0);
  for (int s = 0; s < NSTAGE; ++s) {
    const int cur = s & 1;
    if (s + 1 < NSTAGE) {
      issue_stage(s + 1, cur ^ 1);
      asm volatile("s_wait_asynccnt 0x4" ::: "memory");  // current stage done
    } else {
      asm volatile("s_wait_asynccnt 0x0" ::: "memory");
    }
    __syncthreads();
#pragma unroll
    for (int tt = 0; tt < 2; ++tt) {
      const int col = s * STAGE_COLS + tt * 16 + lane15;
      v8f acc = tile_gemm(cur, tt);
      const float sqc = sqn[col];
      const int   yc  = y[col];
#pragma unroll
      for (int r = 0; r < 8; ++r) {
        // clamp to >= 0 for free: negative floats have int bits < 0
        int d2i = __float_as_int(sqr[r] + sqc - 2.f * acc[r]);
        unsigned d2u  = (unsigned)(d2i < 0 ? 0 : d2i);
        unsigned cand = (yc == yr[r]) ? d2u : HIGH_U;
        insert4u(cand, s0[r], s1[r], s2[r], s3[r]);
      }
    }
    __syncthreads();
  }

  // cross-lane merge of sorted top-4 lists over the 16 column-lanes of each
  // half: 4-stage shfl_xor merge network; afterwards every lane holds the
  // global top-4 for its rows -> anchors live in registers, no LDS needed.
#pragma unroll
  for (int off = 1; off <= 8; off <<= 1) {
#pragma unroll
    for (int r = 0; r < 8; ++r) {
      unsigned t0 = __shfl_xor(s0[r], off, 16);
      unsigned t1 = __shfl_xor(s1[r], off, 16);
      unsigned t2 = __shfl_xor(s2[r], off, 16);
      unsigned t3 = __shfl_xor(s3[r], off, 16);
      insert4u(t0, s0[r], s1[r], s2[r], s3[r]);
      insert4u(t1, s0[r], s1[r], s2[r], s3[r]);
      insert4u(t2, s0[r], s1[r], s2[r], s3[r]);
      insert4u(t3, s0[r], s1[r], s2[r], s3[r]);
    }
  }
  // anchor = (K+1)-th smallest incl. self@0 = s3[r]

  // =============== Pass B: count all points with d^2 <= anchor =============
  int cnt[8];
#pragma unroll
  for (int r = 0; r < 8; ++r) cnt[r] = 0;

  issue_stage(0, 0);
  for (int s = 0; s < NSTAGE; ++s) {
    const int cur = s & 1;
    if (s + 1 < NSTAGE) {
      issue_stage(s + 1, cur ^ 1);
      asm volatile("s_wait_asynccnt 0x4" ::: "memory");
    } else {
      asm volatile("s_wait_asynccnt 0x0" ::: "memory");
    }
    __syncthreads();
#pragma unroll
    for (int tt = 0; tt < 2; ++tt) {
      const int col = s * STAGE_COLS + tt * 16 + lane15;
      v8f acc = tile_gemm(cur, tt);
      const float sqc = sqn[col];
#pragma unroll
      for (int r = 0; r < 8; ++r) {
        int d2i = __float_as_int(sqr[r] + sqc - 2.f * acc[r]);
        unsigned d2u = (unsigned)(d2i < 0 ? 0 : d2i);
        cnt[r] += (d2u <= s3[r]) ? 1 : 0;
      }
    }
    __syncthreads();
  }

  // sum counts across the 16 column-lanes of each half (wave32 shuffles)
#pragma unroll
  for (int r = 0; r < 8; ++r) {
#pragma unroll
    for (int off = 8; off >= 1; off >>= 1)
      cnt[r] += __shfl_xor(cnt[r], off, 16);
  }

  // every lane in a half now holds the 8 row counts for rows r + 8*halfSel
  float psum = 0.f;
#pragma unroll
  for (int r = 0; r < 8; ++r) {
    int m_i = cnt[r] - 1;            // exclude self
    if (m_i < 1) m_i = 1;
    psum += digammaf_dev((float)m_i);
  }
  psum += __shfl_xor(psum, 16, 32);  // combine the two halves
  if (lane == 0) partials[waveId] = psum;
}

__global__ void finalize_kernel(const int* __restrict__ cnts,
                                const float* __restrict__ partials,
                                float* __restrict__ out) {
  if (blockIdx.x != 0 || threadIdx.x != 0) return;
  float tot = 0.f;
  for (int i = 0; i < N_PTS / 16; ++i) tot += partials[i];   // fixed order
  float avg_m = tot / (float)N_PTS;
  float avgNx = 0.f;
  for (int c = 0; c < NCLS; ++c) {
    float nx = (float)cnts[c];
    if (nx > 0.f) avgNx += (nx / (float)N_PTS) * digammaf_dev(nx);
  }
  float mi = digammaf_dev((float)N_PTS) - avgNx
           + digammaf_dev((float)KNN) - avg_m;
  out[0] = fmaxf(mi / logf(2.f), 0.f);
}

extern "C" void kernel_launch(void* const* d_in, const int* in_sizes, int n_in,
                              void* d_out, int out_size, void* d_ws, size_t ws_size,
                              hipStream_t stream) {
  const float* X = (const float*)d_in[0];
  const int*   y = (const int*)d_in[1];

  // workspace layout: [ sqn: 8192 f32 | class counts: 10 i32 | pad | partials: 512 f32 ]
  float* sqn      = (float*)d_ws;
  int*   cnts     = (int*)((char*)d_ws + N_PTS * sizeof(float));
  float* partials = (float*)((char*)d_ws + N_PTS * sizeof(float) + 64);

  zero_counts_kernel<<<1, 32, 0, stream>>>(cnts);
  prep_kernel<<<N_PTS / 256, 256, 0, stream>>>(X, y, sqn, cnts);
  // 512 waves (one per 16-row strip) = 64 blocks * 256 threads (8 waves)
  mi_main_kernel<<<(N_PTS / 16) * 32 / 256, 256, 0, stream>>>(X, y, sqn, partials);
  finalize_kernel<<<1, 32, 0, stream>>>(cnts, partials, (float*)d_out);
}